// caps_Conv2d_48515950576311
// MI455X (gfx1250) — compile-verified
//
#include <hip/hip_runtime.h>
#include <hip/hip_bf16.h>

// Problem constants (from reference setup_inputs)
#define BB   4
#define CC   32
#define HH   16
#define WW   16
#define DIN  8
#define OO   32
#define DOUT 16
#define NPAT (CC * 9)          // 288 = C*K*K
#define MDIM (BB * HH * WW)    // 1024 pixels
#define KDIM (NPAT * DIN)      // 2304
#define NDIM (OO * DOUT)       // 512

typedef __attribute__((ext_vector_type(2))) float v2f;
typedef __attribute__((ext_vector_type(8))) float v8f;
typedef __attribute__((ext_vector_type(4))) int   v4i;

#if __has_builtin(__builtin_amdgcn_global_load_async_to_lds_b128) && \
    __has_builtin(__builtin_amdgcn_global_load_async_to_lds_b32)  && \
    __has_builtin(__builtin_amdgcn_s_wait_asynccnt)
#define USE_ASYNC_LDS 1
#else
#define USE_ASYNC_LDS 0
#endif

#if USE_ASYNC_LDS
// builtin signatures (from clang diagnostic / LLVM def "vV4i*1V4i*3IiIi"):
//   b128: (v4i as1*, v4i as3*, imm offset, imm cpol)
//   b32 : (int as1*, int as3*, imm offset, imm cpol)
typedef __attribute__((address_space(1))) v4i g_v4i;
typedef __attribute__((address_space(3))) v4i l_v4i;
typedef __attribute__((address_space(1))) int g_i32;
typedef __attribute__((address_space(3))) int l_i32;

__device__ __forceinline__ void async_g2l_b128(const float* g, float* l) {
    __builtin_amdgcn_global_load_async_to_lds_b128((g_v4i*)g, (l_v4i*)l, 0, 0);
}
__device__ __forceinline__ void async_g2l_b32(const float* g, float* l) {
    __builtin_amdgcn_global_load_async_to_lds_b32((g_i32*)g, (l_i32*)l, 0, 0);
}
#endif

// ---------------------------------------------------------------------------
// Kernel 1: im2col  ->  A[m][k], m = ((b*16+h)*16+w), k = n*8+i, n = c*9+kh*3+kw
// One thread per (m, n): copies 8 contiguous floats (the Din vector).
// ---------------------------------------------------------------------------
__global__ __launch_bounds__(256) void caps_im2col(const float* __restrict__ inp,
                                                   float* __restrict__ A) {
    int idx = blockIdx.x * 256 + threadIdx.x;          // over M * NPAT
    if (idx >= MDIM * NPAT) return;
    int m = idx / NPAT;
    int n = idx - m * NPAT;
    int c  = n / 9;
    int r  = n - c * 9;
    int kh = r / 3;
    int kw = r - kh * 3;
    int b = m >> 8;
    int h = (m >> 4) & 15;
    int w = m & 15;
    int hh = h + kh - 1;
    int ww = w + kw - 1;
    float4 lo = make_float4(0.f, 0.f, 0.f, 0.f), hi = lo;
    if ((unsigned)hh < 16u && (unsigned)ww < 16u) {
        const float* src = inp + ((((size_t)b * CC + c) * HH + hh) * WW + ww) * DIN;
        lo = *(const float4*)(src);
        hi = *(const float4*)(src + 4);
    }
    float* dst = A + (size_t)idx * DIN;                // (m*288+n)*8 == m*2304 + n*8
    *(float4*)(dst)     = lo;
    *(float4*)(dst + 4) = hi;
}

// ---------------------------------------------------------------------------
// Kernel 2: pack W[o,n,j,i] -> Bmat[k][col] row-major, k=n*8+i, col=o*16+j
// ---------------------------------------------------------------------------
__global__ __launch_bounds__(256) void caps_packB(const float* __restrict__ Wt,
                                                  float* __restrict__ Bm) {
    int idx = blockIdx.x * 256 + threadIdx.x;          // over KDIM * NDIM
    if (idx >= KDIM * NDIM) return;
    int k   = idx / NDIM;
    int col = idx - k * NDIM;
    int n = k >> 3, i = k & 7;
    int o = col >> 4, j = col & 15;
    Bm[idx] = Wt[(((size_t)o * NPAT + n) * DOUT + j) * DIN + i];
}

// ---------------------------------------------------------------------------
// Kernel 3: WMMA GEMM (f32 16x16x4) + squash epilogue + transposed store.
// Block = 256 threads = 8 waves arranged 4(M) x 2(N); block tile 64x32.
// Grid = (M/64, N/32) = (16, 16).
// LDS staging uses the CDNA5 async copy engine (ASYNCcnt) when available.
// ---------------------------------------------------------------------------
__global__ __launch_bounds__(256) void caps_gemm_squash(const float* __restrict__ A,
                                                        const float* __restrict__ Bm,
                                                        float* __restrict__ out) {
    // +68 stride: float4/b128 LDS writes stay 16B aligned (68*4=272),
    // and per-lane b64 fragment reads hit distinct LDS banks (stride%64 = 4).
    __shared__ float As[64][68];
    __shared__ float Bs[32][68];   // stored transposed: Bs[col][k]
    __shared__ float Cs[8][16][17];

    const int t    = threadIdx.x;
    const int wave = t >> 5;
    const int lane = t & 31;
    const int lm   = lane & 15;
    const int hi   = lane >> 4;        // half-wave select
    const int msub = wave & 3;         // 0..3
    const int nsub = wave >> 2;        // 0..1
    const int mBase = blockIdx.x * 64;
    const int nBase = blockIdx.y * 32;

    // cooperative-load mappings
    const int arow  = t >> 4;          // 0..15
    const int acol4 = (t & 15) << 2;   // 0,4,...,60
    const int bk    = t >> 5;          // 0..7
    const int bcol  = t & 31;          // 0..31

    v8f acc = {0.f, 0.f, 0.f, 0.f, 0.f, 0.f, 0.f, 0.f};

    for (int kc = 0; kc < KDIM; kc += 64) {
        // ---- stage A chunk: 64 rows x 64 k ----
#if USE_ASYNC_LDS
#pragma unroll
        for (int it = 0; it < 4; ++it) {
            int row = arow + it * 16;
            async_g2l_b128(&A[(size_t)(mBase + row) * KDIM + kc + acol4],
                           &As[row][acol4]);
        }
        // ---- stage B chunk transposed: Bs[col][k] ----
#pragma unroll
        for (int it = 0; it < 8; ++it) {
            int kk = bk + it * 8;
            async_g2l_b32(&Bm[(size_t)(kc + kk) * NDIM + nBase + bcol],
                          &Bs[bcol][kk]);
        }
        __builtin_amdgcn_s_wait_asynccnt(0);   // own async LDS writes done
#else
#pragma unroll
        for (int it = 0; it < 4; ++it) {
            int row = arow + it * 16;
            float4 v = *(const float4*)&A[(size_t)(mBase + row) * KDIM + kc + acol4];
            *(float4*)&As[row][acol4] = v;
        }
#pragma unroll
        for (int it = 0; it < 8; ++it) {
            int kk = bk + it * 8;
            Bs[bcol][kk] = Bm[(size_t)(kc + kk) * NDIM + nBase + bcol];
        }
#endif
        // prefetch next A chunk (emits global_prefetch_b8)
        if (kc + 64 < KDIM)
            __builtin_prefetch(&A[(size_t)(mBase + arow) * KDIM + kc + 64 + acol4], 0, 3);
        __syncthreads();

        // ---- 16 WMMAs per chunk: D(16x16) += A(16x4) * B(4x16) ----
        // A frag: lane M = lane%16; VGPR0 = K+{0|2}, VGPR1 = K+{1|3}  -> one b64 read
        // B frag: lane N = lane%16; VGPR0 = K+{0|2}, VGPR1 = K+{1|3}  -> one b64 read (Bs is [col][k])
#pragma unroll
        for (int kk = 0; kk < 64; kk += 4) {
            v2f av = *(const v2f*)&As[msub * 16 + lm][kk + 2 * hi];
            v2f bv = *(const v2f*)&Bs[nsub * 16 + lm][kk + 2 * hi];
            acc = __builtin_amdgcn_wmma_f32_16x16x4_f32(
                false, av, false, bv, (short)0, acc, false, false);
        }
        __syncthreads();
    }

    // ---- epilogue: stage C tile, squash over j (the 16 columns = Dout) ----
    // C/D layout: VGPR r -> M = r + 8*hi, N = lane%16
#pragma unroll
    for (int r = 0; r < 8; ++r)
        Cs[wave][r + 8 * hi][lm] = acc[r];
    __syncthreads();

    // each lane owns row lm of its wave's tile; both halves compute the row
    // norm, each half writes its 8 output floats (aligned float4 stores).
    const int m = mBase + msub * 16 + lm;              // pixel index
    const int o = (nBase + nsub * 16) >> 4;            // output capsule channel
    float row[16];
    float sq = 0.f;
#pragma unroll
    for (int j = 0; j < 16; ++j) {
        row[j] = Cs[wave][lm][j];
        sq += row[j] * row[j];
    }
    const float scale = sq / ((1.f + sq) * (__builtin_sqrtf(sq) + 1e-8f));
    const int b = m >> 8;
    const int h = (m >> 4) & 15;
    const int w = m & 15;
    float* dst = out + ((((size_t)b * OO + o) * HH + h) * WW + w) * DOUT + hi * 8;
    float4 o0 = make_float4(row[hi * 8 + 0] * scale, row[hi * 8 + 1] * scale,
                            row[hi * 8 + 2] * scale, row[hi * 8 + 3] * scale);
    float4 o1 = make_float4(row[hi * 8 + 4] * scale, row[hi * 8 + 5] * scale,
                            row[hi * 8 + 6] * scale, row[hi * 8 + 7] * scale);
    *(float4*)(dst)     = o0;
    *(float4*)(dst + 4) = o1;
}

// ---------------------------------------------------------------------------
// Launch: ws layout = [ A : 1024*2304 f32 | Bmat : 2304*512 f32 ]  (~14.2 MB)
// ---------------------------------------------------------------------------
extern "C" void kernel_launch(void* const* d_in, const int* in_sizes, int n_in,
                              void* d_out, int out_size, void* d_ws, size_t ws_size,
                              hipStream_t stream) {
    const float* inp = (const float*)d_in[0];
    const float* Wt  = (const float*)d_in[1];
    float* out  = (float*)d_out;
    float* Amat = (float*)d_ws;
    float* Bmat = Amat + (size_t)MDIM * KDIM;

    caps_im2col<<<(MDIM * NPAT + 255) / 256, 256, 0, stream>>>(inp, Amat);
    caps_packB<<<(KDIM * NDIM + 255) / 256, 256, 0, stream>>>(Wt, Bmat);
    caps_gemm_squash<<<dim3(MDIM / 64, NDIM / 32), 256, 0, stream>>>(Amat, Bmat, out);
}